// GATLayer_35296041239291
// MI455X (gfx1250) — compile-verified
//
#include <hip/hip_runtime.h>

typedef __attribute__((ext_vector_type(2))) float v2f;
typedef __attribute__((ext_vector_type(8))) float v8f;

#define N_NODES 4096
#define D_IN    256
#define D_OUT   128
#define H_HEADS 4
#define NTILES  (N_NODES / 16)
#define NWORDS  (N_NODES / 32)
#define NEG_BIG (-3.0e38f)
#define SLOPE   0.2f

// hB layout: element ((head*NTILES + jt)*4 + i) * 512 + o*4 + c
//   where node n = jt*16 + i*4 + c  (i = K-step, c = row within K-quad)
// A lane's B operand (b.x = h[4i+jb][o], b.y = h[4i+jb+1][o]) is then one
// aligned float2 load; a wave's 32 loads for one o-tile cover one contiguous
// 256B region.

// ---------------------------------------------------------------------------
// Kernel 1: h = x @ W per head (f32 WMMA 16x16x4), stored in hB swizzle.
// One wave -> 16x128 strip for one head.
// ---------------------------------------------------------------------------
__global__ __launch_bounds__(32) void gat_h_kernel(const float* __restrict__ x,
                                                   const float* __restrict__ W,
                                                   float* __restrict__ hB) {
    const int jt   = blockIdx.x;        // 16-node tile index
    const int head = blockIdx.y;
    const int lane = threadIdx.x;
    const int row  = lane & 15;
    const int hi   = lane >> 4;
    const int kb   = hi * 2;

    const float* xrow = x + (jt * 16 + row) * D_IN;
    const float* Wh   = W + head * D_IN * D_OUT;

    v8f acc[8] = {};

    for (int k0 = 0; k0 < D_IN; k0 += 4) {
        v2f a = *(const v2f*)(xrow + k0 + kb);      // x[row][k0+kb, k0+kb+1]
#pragma unroll
        for (int ot = 0; ot < 8; ++ot) {
            const float* wp = Wh + (k0 + kb) * D_OUT + ot * 16 + row;
            v2f b;
            b.x = wp[0];
            b.y = wp[D_OUT];
            acc[ot] = __builtin_amdgcn_wmma_f32_16x16x4_f32(
                false, a, false, b, (short)0, acc[ot], false, false);
        }
    }

    // Store into hB swizzle: local row lr = r + hi*8 -> (i = lr>>2, c = lr&3)
    float* hb = hB + ((size_t)(head * NTILES + jt)) * (4 * D_OUT * 4);
#pragma unroll
    for (int ot = 0; ot < 8; ++ot)
#pragma unroll
        for (int r = 0; r < 8; ++r) {
            int lr = r + hi * 8;
            hb[(lr >> 2) * (D_OUT * 4) + (ot * 16 + row) * 4 + (lr & 3)] =
                acc[ot][r];
        }
}

// ---------------------------------------------------------------------------
// Kernel 2: s_i / s_j projections (tiny).
// ---------------------------------------------------------------------------
__global__ __launch_bounds__(128) void gat_s_kernel(const float* __restrict__ hB,
                                                    const float* __restrict__ a,
                                                    float* __restrict__ sI,
                                                    float* __restrict__ sJ) {
    int idx = blockIdx.x * blockDim.x + threadIdx.x;   // head*N + n
    if (idx >= H_HEADS * N_NODES) return;
    int head = idx / N_NODES;
    int n    = idx - head * N_NODES;
    int jt = n >> 4, i = (n & 15) >> 2, c = n & 3;
    const float* hp = hB + ((size_t)(head * NTILES + jt) * 4 + i) * (D_OUT * 4) + c;
    const float* a1 = a + head * 2 * D_OUT;
    const float* a2 = a1 + D_OUT;
    float s1 = 0.f, s2 = 0.f;
#pragma unroll 4
    for (int o = 0; o < D_OUT; ++o) {
        float v = hp[o * 4];
        s1 += v * a1[o];
        s2 += v * a2[o];
    }
    sI[idx] = s1;
    sJ[idx] = s2;
}

// ---------------------------------------------------------------------------
// Kernel 3: prepass. One wave per row (all 4 heads):
//   - pack mask bits via __ballot (2 MB, L2-resident for main kernel)
//   - row max m_h, then row sum l_h, write z = m + log(l)  (log-sum-exp)
// ---------------------------------------------------------------------------
__global__ __launch_bounds__(128) void gat_prep_kernel(const int* __restrict__ A,
                                                       const float* __restrict__ sI,
                                                       const float* __restrict__ sJ,
                                                       float* __restrict__ z,
                                                       unsigned* __restrict__ maskbits) {
    __shared__ unsigned smask[4][NWORDS];   // per-wave private rows, no barriers
    const int wid  = threadIdx.x >> 5;
    const int lane = threadIdx.x & 31;
    const int row  = blockIdx.x * 4 + wid;

    const int* Arow = A + (size_t)row * N_NODES;
    float si[H_HEADS];
#pragma unroll
    for (int h = 0; h < H_HEADS; ++h) si[h] = sI[h * N_NODES + row];

    float m[H_HEADS] = {NEG_BIG, NEG_BIG, NEG_BIG, NEG_BIG};

    // pass A: ballot mask words + per-lane running max (branchless)
    for (int j = lane; j < N_NODES; j += 32) {
        int av = Arow[j];
        bool msk = (av > 0) || (j == row);
        unsigned bal = (unsigned)__ballot(msk);
        if (lane == 0) {
            smask[wid][j >> 5] = bal;
            maskbits[(size_t)row * NWORDS + (j >> 5)] = bal;
        }
#pragma unroll
        for (int h = 0; h < H_HEADS; ++h) {
            float e = si[h] + sJ[h * N_NODES + j];
            e = (e > 0.f) ? e : SLOPE * e;
            m[h] = fmaxf(m[h], msk ? e : NEG_BIG);
        }
    }
#pragma unroll
    for (int h = 0; h < H_HEADS; ++h)
#pragma unroll
        for (int off = 16; off > 0; off >>= 1)
            m[h] = fmaxf(m[h], __shfl_xor(m[h], off));

    // pass B: row sums with fixed max
    float l[H_HEADS] = {0.f, 0.f, 0.f, 0.f};
    for (int j = lane; j < N_NODES; j += 32) {
        bool msk = (smask[wid][j >> 5] >> lane) & 1u;
#pragma unroll
        for (int h = 0; h < H_HEADS; ++h) {
            float e = si[h] + sJ[h * N_NODES + j];
            e = (e > 0.f) ? e : SLOPE * e;
            l[h] += msk ? __expf(e - m[h]) : 0.f;
        }
    }
#pragma unroll
    for (int h = 0; h < H_HEADS; ++h)
#pragma unroll
        for (int off = 16; off > 0; off >>= 1)
            l[h] += __shfl_xor(l[h], off);

    if (lane == 0)
#pragma unroll
        for (int h = 0; h < H_HEADS; ++h)
            z[h * N_NODES + row] = m[h] + __logf(l[h]);
}

// ---------------------------------------------------------------------------
// Kernel 4: fused attention @ h. Block = 4 waves; wave w = head w, 16 rows.
// P = exp(e - z_i) produced directly in WMMA A-operand layout; no softmax
// bookkeeping in the hot loop. B operands are single b64 loads from hB.
// ---------------------------------------------------------------------------
__global__ __launch_bounds__(128) void gat_attn_kernel(const float* __restrict__ hB,
                                                       const unsigned* __restrict__ maskbits,
                                                       const float* __restrict__ sI,
                                                       const float* __restrict__ sJ,
                                                       const float* __restrict__ z,
                                                       float* __restrict__ out) {
    const int i0   = blockIdx.x * 16;
    const int head = threadIdx.x >> 5;
    const int lane = threadIdx.x & 31;
    const int row  = lane & 15;
    const int hi   = lane >> 4;
    const int jb   = hi * 2;

    const int gi = i0 + row;
    const float si = sI[head * N_NODES + gi];
    const float zi = z[head * N_NODES + gi];
    const float* sjh  = sJ + head * N_NODES;
    const unsigned* mrow = maskbits + (size_t)gi * NWORDS;
    const float* hBh = hB + (size_t)head * NTILES * (4 * D_OUT * 4);
    const int laneoff = row * 4 + jb;

    v8f acc[8] = {};

    for (int j0 = 0; j0 < N_NODES; j0 += 32) {
        unsigned mw = mrow[j0 >> 5];
#pragma unroll
        for (int halfT = 0; halfT < 2; ++halfT) {
            const int jt0 = j0 + halfT * 16;
            const unsigned mb = mw >> (halfT * 16);

            // normalized probabilities, directly in A-operand order
            float p8[8];
#pragma unroll
            for (int i = 0; i < 4; ++i) {
                int jl = 4 * i + jb;
                v2f sjv = *(const v2f*)(sjh + jt0 + jl);
                float e0 = si + sjv.x; e0 = (e0 > 0.f) ? e0 : SLOPE * e0;
                float e1 = si + sjv.y; e1 = (e1 > 0.f) ? e1 : SLOPE * e1;
                p8[2 * i]     = ((mb >> jl) & 1u)       ? __expf(e0 - zi) : 0.f;
                p8[2 * i + 1] = ((mb >> (jl + 1)) & 1u) ? __expf(e1 - zi) : 0.f;
            }

            // acc += P(16x16) @ h_tile(16x128)
            const float* hbt = hBh + (size_t)(jt0 >> 4) * (4 * D_OUT * 4) + laneoff;
#pragma unroll
            for (int i = 0; i < 4; ++i) {
                v2f ap;
                ap.x = p8[2 * i];
                ap.y = p8[2 * i + 1];
                const float* hk = hbt + i * (D_OUT * 4);
#pragma unroll
                for (int ot = 0; ot < 8; ++ot) {
                    v2f b = *(const v2f*)(hk + ot * 64);
                    acc[ot] = __builtin_amdgcn_wmma_f32_16x16x4_f32(
                        false, ap, false, b, (short)0, acc[ot], false, false);
                }
            }
        }
    }

    // store (already normalized): out[n][head*128 + o]
    float* orow = out + (size_t)i0 * (H_HEADS * D_OUT) + head * D_OUT;
#pragma unroll
    for (int ot = 0; ot < 8; ++ot)
#pragma unroll
        for (int r = 0; r < 8; ++r)
            orow[(r + hi * 8) * (H_HEADS * D_OUT) + ot * 16 + row] = acc[ot][r];
}

// ---------------------------------------------------------------------------
extern "C" void kernel_launch(void* const* d_in, const int* in_sizes, int n_in,
                              void* d_out, int out_size, void* d_ws, size_t ws_size,
                              hipStream_t stream) {
    const float* x = (const float*)d_in[0];
    const int*   A = (const int*)d_in[1];
    const float* W = (const float*)d_in[2];
    const float* a = (const float*)d_in[3];
    float* out = (float*)d_out;

    float* hB = (float*)d_ws;                                   // 8 MB
    float* sI = hB + (size_t)H_HEADS * N_NODES * D_OUT;
    float* sJ = sI + (size_t)H_HEADS * N_NODES;
    float* z  = sJ + (size_t)H_HEADS * N_NODES;
    unsigned* maskbits = (unsigned*)(z + (size_t)H_HEADS * N_NODES);  // 2 MB

    gat_h_kernel<<<dim3(NTILES, H_HEADS), 32, 0, stream>>>(x, W, hB);
    gat_s_kernel<<<(H_HEADS * N_NODES) / 128, 128, 0, stream>>>(hB, a, sI, sJ);
    gat_prep_kernel<<<N_NODES / 4, 128, 0, stream>>>(A, sI, sJ, z, maskbits);
    gat_attn_kernel<<<NTILES, 128, 0, stream>>>(hB, maskbits, sI, sJ, z, out);
}